// S4Layer_33621003993784
// MI455X (gfx1250) — compile-verified
//
#include <hip/hip_runtime.h>

// ---------------------------------------------------------------------------
// S4 layer for MI455X (gfx1250, wave32).
//   u:(4,4096,1024) f32 -> y:(4,4096,1024) f32
// Two fp32 WMMA GEMMs (16x16x4) around a chunked parallel complex scan.
// GEMM tiles: 32 rows x 128 cols per 256-thread block (8 waves, 2 acc/wave),
// double-buffered LDS fed by gfx1250 async global->LDS loads when available.
// ---------------------------------------------------------------------------

typedef __attribute__((ext_vector_type(2))) float v2f;
typedef __attribute__((ext_vector_type(8))) float v8f;

#define T_LEN   4096
#define BATCH   4
#define HDIM    1024
#define NSTATE  64
#define NCOL    (2 * NSTATE)          // 128 (re/im interleaved columns)
#define MROWS   (BATCH * T_LEN)       // 16384
#define MTILE   32                    // rows per block
#define KC      32                    // K-chunk per LDS stage
#define APITCH  (KC + 2)              // 34: conflict-free, rows 8B-aligned
#define NCH     64                    // scan chunks
#define CLEN    64                    // timesteps per scan chunk

// ---- workspace layout (float offsets) -------------------------------------
#define OFF_ABAR 0u
#define OFF_COEF (OFF_ABAR + 2u * NSTATE)                       // 128
#define OFF_B1   (OFF_COEF + 2u * NSTATE)                       // 256
#define OFF_WT   (OFF_B1 + (size_t)HDIM * NCOL)                 // +131072
#define OFF_BU   (OFF_WT + (size_t)NCOL * HDIM)                 // +131072
#define OFF_XS   (OFF_BU + (size_t)MROWS * NCOL)                // +2097152
#define OFF_CS   (OFF_XS + (size_t)MROWS * NCOL)                // +2097152
#define OFF_CI   (OFF_CS + (size_t)NCH * BATCH * NSTATE * 2)    // +32768

// ---- gfx1250 async global->LDS path ---------------------------------------
#if defined(__AMDGCN__) && __has_builtin(__builtin_amdgcn_global_load_async_to_lds_b128) && \
    __has_builtin(__builtin_amdgcn_global_load_async_to_lds_b64)
#define USE_ASYNC_LDS 1
#else
#define USE_ASYNC_LDS 0
#endif

#if __has_builtin(__builtin_amdgcn_s_wait_asynccnt)
#define WAIT_ASYNC0() __builtin_amdgcn_s_wait_asynccnt(0)
#else
#define WAIT_ASYNC0() asm volatile("s_wait_asynccnt 0x0" ::: "memory")
#endif

#if USE_ASYNC_LDS
// Types exactly as in the builtin's signature (from the compiler diagnostic):
//   param0: AS1 pointer to int __attribute__((vector_size(8/16)))
//   param1: AS3 pointer
typedef int i2vec __attribute__((__vector_size__(2 * sizeof(int))));
typedef int i4vec __attribute__((__vector_size__(4 * sizeof(int))));
typedef __attribute__((address_space(1))) i2vec* gp_i2;
typedef __attribute__((address_space(1))) i4vec* gp_i4;
typedef __attribute__((address_space(3))) i2vec* lp_i2;
typedef __attribute__((address_space(3))) i4vec* lp_i4;

__device__ __forceinline__ gp_i2 glb2(const void* p) { return (gp_i2)(uintptr_t)p; }
__device__ __forceinline__ gp_i4 glb4(const void* p) { return (gp_i4)(uintptr_t)p; }
// flat LDS address: low 32 bits are the LDS offset
__device__ __forceinline__ lp_i2 lds2(void* p) { return (lp_i2)(uintptr_t)p; }
__device__ __forceinline__ lp_i4 lds4(void* p) { return (lp_i4)(uintptr_t)p; }
#endif

// ---------------------------------------------------------------------------
// Discretization scalars: A_bar = exp(A*step), coef = (A_bar-1)/A
// ---------------------------------------------------------------------------
__global__ __launch_bounds__(64) void k_scalars(const float* __restrict__ log_A_real,
                                                const float* __restrict__ A_imag,
                                                const float* __restrict__ log_step,
                                                float* __restrict__ abar,
                                                float* __restrict__ coef) {
    int n = threadIdx.x;
    float step = __expf(log_step[0]);
    float Ar = -__expf(log_A_real[n]);
    float Ai = A_imag[n];
    float e   = __expf(Ar * step);
    float abr = e * __cosf(Ai * step);
    float abi = e * __sinf(Ai * step);
    float nr = abr - 1.0f, ni = abi;
    float den = Ar * Ar + Ai * Ai;
    float inv = 1.0f / den;
    float cr = (nr * Ar + ni * Ai) * inv;
    float ci = (ni * Ar - nr * Ai) * inv;
    abar[2 * n] = abr;  abar[2 * n + 1] = abi;
    coef[2 * n] = cr;   coef[2 * n + 1] = ci;
}

// B1[h][2n] = Re(coef[n])*B[n,h] ; B1[h][2n+1] = Im(coef[n])*B[n,h]   (HDIM x 128)
__global__ __launch_bounds__(256) void k_build_b1(const float* __restrict__ Bw,
                                                  const float* __restrict__ coef,
                                                  float* __restrict__ B1) {
    int idx = blockIdx.x * 256 + threadIdx.x;        // over NSTATE*HDIM
    int n = idx >> 10;                                // /HDIM
    int h = idx & (HDIM - 1);
    float b = Bw[n * HDIM + h];
    B1[(size_t)h * NCOL + 2 * n]     = coef[2 * n] * b;
    B1[(size_t)h * NCOL + 2 * n + 1] = coef[2 * n + 1] * b;
}

// Wt[2n][h] = C[h,n,0] ; Wt[2n+1][h] = -C[h,n,1]    (128 x HDIM)
__global__ __launch_bounds__(256) void k_build_wt(const float* __restrict__ C,
                                                  float* __restrict__ Wt) {
    int idx = blockIdx.x * 256 + threadIdx.x;        // over HDIM*NSTATE
    int h = idx >> 6;
    int n = idx & (NSTATE - 1);
    Wt[(size_t)(2 * n) * HDIM + h]     =  C[((size_t)h * NSTATE + n) * 2 + 0];
    Wt[(size_t)(2 * n + 1) * HDIM + h] = -C[((size_t)h * NSTATE + n) * 2 + 1];
}

// ---------------------------------------------------------------------------
// WMMA GEMM: Out[M x Ntot] = A[M x K] * Bm[K x Ntot]  (+ D[h]*u epilogue)
// Block = 256 threads = 8 wave32, tile = 32 rows x 128 cols.
// Each wave owns two v_wmma_f32_16x16x4_f32 accumulators sharing one B frag.
// ---------------------------------------------------------------------------
__global__ __launch_bounds__(256) void k_gemm(const float* __restrict__ A, int lda,
                                              const float* __restrict__ Bm, int ldb,
                                              float* __restrict__ Out, int ldo,
                                              int K,
                                              const float* __restrict__ uEp,
                                              const float* __restrict__ dEp,
                                              int useEpi) {
    __shared__ __align__(16) float As[2][MTILE][APITCH];
    __shared__ __align__(16) float Bs[2][KC][NCOL];

    const int tid  = threadIdx.x;
    const int lane = tid & 31;
    const int wave = tid >> 5;                 // 0..7
    const int rowBase = blockIdx.x * MTILE;
    const int colBase = blockIdx.y * NCOL;
    const int m    = lane & 15;                // A row / D row-lane
    const int koff = (lane >> 4) << 1;         // 0 or 2 (K sub-offset per ISA layout)
    const int wcol = wave * 16 + m;            // local column (0..127)

    v8f acc0 = {};
    v8f acc1 = {};
    const int nch = K / KC;

#if USE_ASYNC_LDS
    // ---- async copy assignments ----
    // A: 32x32 f32 chunk as 512 B64 transfers -> 2 per thread
    const int aRowA[2] = { tid >> 4, (tid + 256) >> 4 };
    const int aColA    = (tid & 15) << 1;
    // B: 32x128 f32 chunk as 1024 B128 transfers -> 4 per thread
    int bRow[4], bC4[4];
#pragma unroll
    for (int j = 0; j < 4; ++j) {
        int idx4 = tid + j * 256;
        bRow[j] = idx4 >> 5;
        bC4[j]  = (idx4 & 31) << 2;
    }

    // preload chunk 0
    {
#pragma unroll
        for (int j = 0; j < 2; ++j) {
            const int r = aRowA[j];
            __builtin_amdgcn_global_load_async_to_lds_b64(
                glb2(A + (size_t)(rowBase + r) * lda + aColA),
                lds2(&As[0][r][aColA]), 0, 0);
        }
#pragma unroll
        for (int j = 0; j < 4; ++j) {
            __builtin_amdgcn_global_load_async_to_lds_b128(
                glb4(Bm + (size_t)bRow[j] * ldb + colBase + bC4[j]),
                lds4(&Bs[0][bRow[j]][bC4[j]]), 0, 0);
        }
        WAIT_ASYNC0();
    }
    __syncthreads();

    for (int ch = 0; ch < nch; ++ch) {
        const int cur = ch & 1;
        const bool hasNext = (ch + 1) < nch;

        if (hasNext) {                         // async-stage next chunk into other buffer
            const int nxt = cur ^ 1;
            const int k0 = (ch + 1) * KC;
#pragma unroll
            for (int j = 0; j < 2; ++j) {
                const int r = aRowA[j];
                __builtin_amdgcn_global_load_async_to_lds_b64(
                    glb2(A + (size_t)(rowBase + r) * lda + k0 + aColA),
                    lds2(&As[nxt][r][aColA]), 0, 0);
            }
#pragma unroll
            for (int j = 0; j < 4; ++j) {
                __builtin_amdgcn_global_load_async_to_lds_b128(
                    glb4(Bm + (size_t)(k0 + bRow[j]) * ldb + colBase + bC4[j]),
                    lds4(&Bs[nxt][bRow[j]][bC4[j]]), 0, 0);
            }
            if (ch + 2 < nch)
                __builtin_prefetch(Bm + (size_t)((ch + 2) * KC) * ldb + colBase, 0, 1);
        }

#pragma unroll
        for (int kk = 0; kk < KC; kk += 4) {
            v2f a0, a1, bv;
            a0.x = As[cur][m][kk + koff];
            a0.y = As[cur][m][kk + koff + 1];
            a1.x = As[cur][m + 16][kk + koff];
            a1.y = As[cur][m + 16][kk + koff + 1];
            bv.x = Bs[cur][kk + koff][wcol];
            bv.y = Bs[cur][kk + koff + 1][wcol];
            acc0 = __builtin_amdgcn_wmma_f32_16x16x4_f32(false, a0, false, bv, (short)0, acc0, false, false);
            acc1 = __builtin_amdgcn_wmma_f32_16x16x4_f32(false, a1, false, bv, (short)0, acc1, false, false);
        }

        if (hasNext) WAIT_ASYNC0();            // our copies done before the barrier
        __syncthreads();
    }
#else
    // ---- fallback: VGPR-staged double buffering ----
    const int aRow0 = tid >> 5;                // 0..7 (+8/+16/+24 groups)
    const int aCol  = tid & 31;
    int bRow[4], bC4[4];
#pragma unroll
    for (int j = 0; j < 4; ++j) {
        int idx4 = tid + j * 256;
        bRow[j] = idx4 >> 5;
        bC4[j]  = (idx4 & 31) << 2;
    }
    float ra0, ra1, ra2, ra3;
    float4 rb0, rb1, rb2, rb3;
    {
        ra0 = A[(size_t)(rowBase + aRow0) * lda + aCol];
        ra1 = A[(size_t)(rowBase + aRow0 + 8) * lda + aCol];
        ra2 = A[(size_t)(rowBase + aRow0 + 16) * lda + aCol];
        ra3 = A[(size_t)(rowBase + aRow0 + 24) * lda + aCol];
        rb0 = *(const float4*)(Bm + (size_t)bRow[0] * ldb + colBase + bC4[0]);
        rb1 = *(const float4*)(Bm + (size_t)bRow[1] * ldb + colBase + bC4[1]);
        rb2 = *(const float4*)(Bm + (size_t)bRow[2] * ldb + colBase + bC4[2]);
        rb3 = *(const float4*)(Bm + (size_t)bRow[3] * ldb + colBase + bC4[3]);
        As[0][aRow0][aCol]      = ra0;
        As[0][aRow0 + 8][aCol]  = ra1;
        As[0][aRow0 + 16][aCol] = ra2;
        As[0][aRow0 + 24][aCol] = ra3;
        *(float4*)&Bs[0][bRow[0]][bC4[0]] = rb0;
        *(float4*)&Bs[0][bRow[1]][bC4[1]] = rb1;
        *(float4*)&Bs[0][bRow[2]][bC4[2]] = rb2;
        *(float4*)&Bs[0][bRow[3]][bC4[3]] = rb3;
    }
    __syncthreads();

    for (int ch = 0; ch < nch; ++ch) {
        const int cur = ch & 1;
        const bool hasNext = (ch + 1) < nch;
        if (hasNext) {
            const int k0 = (ch + 1) * KC;
            ra0 = A[(size_t)(rowBase + aRow0) * lda + k0 + aCol];
            ra1 = A[(size_t)(rowBase + aRow0 + 8) * lda + k0 + aCol];
            ra2 = A[(size_t)(rowBase + aRow0 + 16) * lda + k0 + aCol];
            ra3 = A[(size_t)(rowBase + aRow0 + 24) * lda + k0 + aCol];
            rb0 = *(const float4*)(Bm + (size_t)(k0 + bRow[0]) * ldb + colBase + bC4[0]);
            rb1 = *(const float4*)(Bm + (size_t)(k0 + bRow[1]) * ldb + colBase + bC4[1]);
            rb2 = *(const float4*)(Bm + (size_t)(k0 + bRow[2]) * ldb + colBase + bC4[2]);
            rb3 = *(const float4*)(Bm + (size_t)(k0 + bRow[3]) * ldb + colBase + bC4[3]);
            if (ch + 2 < nch)
                __builtin_prefetch(Bm + (size_t)((ch + 2) * KC) * ldb + colBase, 0, 1);
        }
#pragma unroll
        for (int kk = 0; kk < KC; kk += 4) {
            v2f a0, a1, bv;
            a0.x = As[cur][m][kk + koff];
            a0.y = As[cur][m][kk + koff + 1];
            a1.x = As[cur][m + 16][kk + koff];
            a1.y = As[cur][m + 16][kk + koff + 1];
            bv.x = Bs[cur][kk + koff][wcol];
            bv.y = Bs[cur][kk + koff + 1][wcol];
            acc0 = __builtin_amdgcn_wmma_f32_16x16x4_f32(false, a0, false, bv, (short)0, acc0, false, false);
            acc1 = __builtin_amdgcn_wmma_f32_16x16x4_f32(false, a1, false, bv, (short)0, acc1, false, false);
        }
        if (hasNext) {
            const int nxt = cur ^ 1;
            As[nxt][aRow0][aCol]      = ra0;
            As[nxt][aRow0 + 8][aCol]  = ra1;
            As[nxt][aRow0 + 16][aCol] = ra2;
            As[nxt][aRow0 + 24][aCol] = ra3;
            *(float4*)&Bs[nxt][bRow[0]][bC4[0]] = rb0;
            *(float4*)&Bs[nxt][bRow[1]][bC4[1]] = rb1;
            *(float4*)&Bs[nxt][bRow[2]][bC4[2]] = rb2;
            *(float4*)&Bs[nxt][bRow[3]][bC4[3]] = rb3;
        }
        __syncthreads();
    }
#endif

    // D-matrix layout: VGPR j -> M = j (lanes 0-15) / j+8 (lanes 16-31), N = lane%16
    const int mAdd = (lane < 16) ? 0 : 8;
    const int c = colBase + wcol;
#pragma unroll
    for (int j = 0; j < 8; ++j) {
        const int r0 = rowBase + mAdd + j;
        const int r1 = r0 + 16;
        float v0 = acc0[j];
        float v1 = acc1[j];
        if (useEpi) {
            v0 += dEp[c] * uEp[(size_t)r0 * ldo + c];
            v1 += dEp[c] * uEp[(size_t)r1 * ldo + c];
        }
        Out[(size_t)r0 * ldo + c] = v0;
        Out[(size_t)r1 * ldo + c] = v1;
    }
}

// ---------------------------------------------------------------------------
// Chunked parallel complex scan: x_t = A_bar * x_{t-1} + Bu_t
// ---------------------------------------------------------------------------
__global__ __launch_bounds__(256) void k_scan_local(const float* __restrict__ Bu,
                                                    const float* __restrict__ abar,
                                                    float* __restrict__ xs,
                                                    float* __restrict__ csum) {
    const int tid = threadIdx.x;              // b*64 + n
    const int b = tid >> 6, n = tid & 63;
    const int chunk = blockIdx.x;
    const float ar = abar[2 * n], ai = abar[2 * n + 1];
    float xr = 0.f, xi = 0.f;
    const int t0 = chunk * CLEN;
#pragma unroll 4
    for (int i = 0; i < CLEN; ++i) {
        const size_t row = (size_t)b * T_LEN + t0 + i;
        const float* p = Bu + row * NCOL + 2 * n;
        float br = p[0], bi = p[1];
        float nr = fmaf(ar, xr, fmaf(-ai, xi, br));
        float ni = fmaf(ar, xi, fmaf(ai, xr, bi));
        xr = nr; xi = ni;
        float* q = xs + row * NCOL + 2 * n;
        q[0] = xr; q[1] = xi;
    }
    csum[(size_t)(chunk * 256 + tid) * 2]     = xr;
    csum[(size_t)(chunk * 256 + tid) * 2 + 1] = xi;
}

__global__ __launch_bounds__(256) void k_scan_carry(const float* __restrict__ abar,
                                                    const float* __restrict__ csum,
                                                    float* __restrict__ cinit) {
    const int tid = threadIdx.x;
    const int n = tid & 63;
    float pr = abar[2 * n], pi = abar[2 * n + 1];
#pragma unroll
    for (int s = 0; s < 6; ++s) {             // A_bar^64 by repeated squaring
        float nr = pr * pr - pi * pi;
        float ni = 2.f * pr * pi;
        pr = nr; pi = ni;
    }
    float cr = 0.f, ci = 0.f;
    for (int c = 0; c < NCH; ++c) {
        cinit[(size_t)(c * 256 + tid) * 2]     = cr;
        cinit[(size_t)(c * 256 + tid) * 2 + 1] = ci;
        float sr = csum[(size_t)(c * 256 + tid) * 2];
        float si = csum[(size_t)(c * 256 + tid) * 2 + 1];
        float nr = fmaf(pr, cr, fmaf(-pi, ci, sr));
        float ni = fmaf(pr, ci, fmaf(pi, cr, si));
        cr = nr; ci = ni;
    }
}

__global__ __launch_bounds__(256) void k_scan_fix(const float* __restrict__ abar,
                                                  const float* __restrict__ cinit,
                                                  float* __restrict__ xs) {
    const int tid = threadIdx.x;
    const int b = tid >> 6, n = tid & 63;
    const int chunk = blockIdx.x;
    float ir = cinit[(size_t)(chunk * 256 + tid) * 2];
    float ii = cinit[(size_t)(chunk * 256 + tid) * 2 + 1];
    if (ir == 0.f && ii == 0.f) return;       // chunk 0 carries nothing
    const float ar = abar[2 * n], ai = abar[2 * n + 1];
    float cr = ar * ir - ai * ii;             // A_bar^(i+1) * init, i = 0
    float ci = ar * ii + ai * ir;
    const int t0 = chunk * CLEN;
#pragma unroll 4
    for (int i = 0; i < CLEN; ++i) {
        const size_t row = (size_t)b * T_LEN + t0 + i;
        float* q = xs + row * NCOL + 2 * n;
        q[0] += cr; q[1] += ci;
        float nr = cr * ar - ci * ai;
        float ni = cr * ai + ci * ar;
        cr = nr; ci = ni;
    }
}

// ---------------------------------------------------------------------------
extern "C" void kernel_launch(void* const* d_in, const int* in_sizes, int n_in,
                              void* d_out, int out_size, void* d_ws, size_t ws_size,
                              hipStream_t stream) {
    const float* u     = (const float*)d_in[0];
    const float* lar   = (const float*)d_in[1];
    const float* aim   = (const float*)d_in[2];
    const float* Bw    = (const float*)d_in[3];
    const float* Cw    = (const float*)d_in[4];
    const float* Dw    = (const float*)d_in[5];
    const float* lstep = (const float*)d_in[6];
    float* out = (float*)d_out;
    float* ws  = (float*)d_ws;

    float* abar = ws + OFF_ABAR;
    float* coef = ws + OFF_COEF;
    float* B1   = ws + OFF_B1;
    float* Wt   = ws + OFF_WT;
    float* Bu   = ws + OFF_BU;
    float* xs   = ws + OFF_XS;
    float* csum = ws + OFF_CS;
    float* cinit= ws + OFF_CI;

    k_scalars<<<1, 64, 0, stream>>>(lar, aim, lstep, abar, coef);
    k_build_b1<<<(NSTATE * HDIM) / 256, 256, 0, stream>>>(Bw, coef, B1);
    k_build_wt<<<(HDIM * NSTATE) / 256, 256, 0, stream>>>(Cw, Wt);

    // GEMM 1: Bu[16384 x 128] = u[16384 x 1024] * B1[1024 x 128]
    dim3 g1(MROWS / MTILE, 1);
    k_gemm<<<g1, 256, 0, stream>>>(u, HDIM, B1, NCOL, Bu, NCOL, HDIM,
                                   nullptr, nullptr, 0);

    // parallel scan over T
    k_scan_local<<<NCH, 256, 0, stream>>>(Bu, abar, xs, csum);
    k_scan_carry<<<1, 256, 0, stream>>>(abar, csum, cinit);
    k_scan_fix<<<NCH, 256, 0, stream>>>(abar, cinit, xs);

    // GEMM 2: y[16384 x 1024] = xs[16384 x 128] * Wt[128 x 1024] + D .* u
    dim3 g2(MROWS / MTILE, HDIM / NCOL);
    k_gemm<<<g2, 256, 0, stream>>>(xs, NCOL, Wt, HDIM, out, HDIM, NCOL,
                                   u, Dw, 1);
}